// Causal_GraphConvolution_80607946211567
// MI455X (gfx1250) — compile-verified
//
#include <hip/hip_runtime.h>

#define KK 2
#define NN 8192
#define IN_F 128
#define OUT_F 64
#define NEG_INF (-9e15f)
#define THREADS 256

typedef float v2f __attribute__((ext_vector_type(2)));
typedef float v8f __attribute__((ext_vector_type(8)));

// ---------------------------------------------------------------------------
// Kernel 1: Wh = x @ weight  via V_WMMA_F32_16X16X4_F32
// Grid: (OUT_F/16, K*N/16), one wave (32 threads) per 16x16 output tile.
// A (16x4 f32): lanes 0-15 hold M=lane, VGPR0=K, VGPR1=K+1; lanes 16-31 K+2,K+3.
// B (4x16 f32): lanes 0-15 hold N=lane, VGPR0=K, VGPR1=K+1; lanes 16-31 K+2,K+3.
// C/D (16x16 f32): VGPR r -> M=r (lanes 0-15) / M=r+8 (lanes 16-31), N=lane&15.
// ---------------------------------------------------------------------------
__global__ __launch_bounds__(32) void k_wh_wmma(const float* __restrict__ x,
                                                const float* __restrict__ w,
                                                float* __restrict__ Wh) {
  const int nt = blockIdx.x;          // 0..3  (col tile)
  const int mt = blockIdx.y;          // 0..1023 (row tile over K*N)
  const int lane = threadIdx.x;
  const int hi = lane >> 4;
  const int ln = lane & 15;
  const int arow = mt * 16 + ln;
  v8f c = {};
  for (int kk = 0; kk < IN_F; kk += 4) {
    const int ka = kk + (hi << 1);
    v2f a = *(const v2f*)(x + (size_t)arow * IN_F + ka);  // K=ka, ka+1 for row arow
    v2f b;
    b.x = w[(ka + 0) * OUT_F + nt * 16 + ln];
    b.y = w[(ka + 1) * OUT_F + nt * 16 + ln];
    c = __builtin_amdgcn_wmma_f32_16x16x4_f32(false, a, false, b,
                                              (short)0, c, false, false);
  }
#pragma unroll
  for (int r = 0; r < 8; ++r) {
    const int orow = mt * 16 + r + (hi << 3);
    Wh[(size_t)orow * OUT_F + nt * 16 + ln] = c[r];
  }
}

// ---------------------------------------------------------------------------
// Kernel 2: Wh1[row] = Wh[row,:] . a[:64],  Wh2[row] = Wh[row,:] . a[64:]
// ---------------------------------------------------------------------------
__global__ __launch_bounds__(THREADS) void k_wh12(const float* __restrict__ Wh,
                                                  const float* __restrict__ a,
                                                  float* __restrict__ Wh1,
                                                  float* __restrict__ Wh2) {
  const int row = blockIdx.x * blockDim.x + threadIdx.x;
  if (row >= KK * NN) return;
  const float* r = Wh + (size_t)row * OUT_F;
  float s1 = 0.f, s2 = 0.f;
#pragma unroll
  for (int o = 0; o < OUT_F; ++o) {
    const float v = r[o];
    s1 = fmaf(v, a[o], s1);
    s2 = fmaf(v, a[OUT_F + o], s2);
  }
  Wh1[row] = s1;
  Wh2[row] = s2;
}

// online-softmax combine: (M,S) <- (M,S) + (m2 with weight s2)
__device__ __forceinline__ void sm_comb(float& M, float& S, float m2, float s2) {
  const float Mn = fmaxf(M, m2);
  S = S * __expf(M - Mn) + s2 * __expf(m2 - Mn);
  M = Mn;
}

// ---------------------------------------------------------------------------
// Kernel 3: per graph row n (both K batches fused -> adj row read once):
//   stats = online softmax over masked relu(Wh1[n]+Wh2[m])
//   h_prime[k,n,:] = sum_{m in nbrs} att[k,n,m] * Wh[k,m,:]
// Deterministic: bitmask scan (32 m per thread) + serial exclusive scan.
// ---------------------------------------------------------------------------
__global__ __launch_bounds__(THREADS) void k_attn(const float* __restrict__ adj,
                                                  const float* __restrict__ Wh,
                                                  const float* __restrict__ Wh1,
                                                  const float* __restrict__ Wh2,
                                                  float* __restrict__ Hp) {
  __shared__ int   s_idx[NN];
  __shared__ float s_w0[1024];
  __shared__ float s_w1[1024];
  __shared__ float sM0[THREADS], sS0[THREADS], sM1[THREADS], sS1[THREADS];
  __shared__ int   s_off[THREADS];
  __shared__ int   s_cnt;
  __shared__ float s_stat[4];

  const int n = blockIdx.x;
  const int tid = threadIdx.x;
  const float w10 = Wh1[n];
  const float w11 = Wh1[NN + n];
  const float* arow = adj + (size_t)n * NN;

  unsigned mask = 0;
  float M0 = NEG_INF, S0 = 0.f, M1 = NEG_INF, S1 = 0.f;
#pragma unroll 4
  for (int i = 0; i < NN / THREADS; ++i) {
    const int m = tid + i * THREADS;            // coalesced across the block
    if (arow[m] > 0.f) {
      mask |= (1u << i);
      sm_comb(M0, S0, fmaxf(w10 + Wh2[m], 0.f), 1.f);
      sm_comb(M1, S1, fmaxf(w11 + Wh2[NN + m], 0.f), 1.f);
    }
  }
  s_off[tid] = __popc(mask);
  sM0[tid] = M0; sS0[tid] = S0; sM1[tid] = M1; sS1[tid] = S1;
  __syncthreads();
  for (int off = THREADS / 2; off > 0; off >>= 1) {
    if (tid < off) {
      sm_comb(sM0[tid], sS0[tid], sM0[tid + off], sS0[tid + off]);
      sm_comb(sM1[tid], sS1[tid], sM1[tid + off], sS1[tid + off]);
    }
    __syncthreads();
  }
  if (tid == 0) {
    s_stat[0] = sM0[0]; s_stat[1] = sS0[0];
    s_stat[2] = sM1[0]; s_stat[3] = sS1[0];
    int run = 0;
    for (int i = 0; i < THREADS; ++i) { const int t = s_off[i]; s_off[i] = run; run += t; }
    s_cnt = run;
  }
  __syncthreads();
  const float rM0 = s_stat[0], rS0 = s_stat[1];
  const float rM1 = s_stat[2], rS1 = s_stat[3];
  const int cnt = s_cnt;
  {
    int off = s_off[tid];
    unsigned mm = mask;
    while (mm) {
      const int i = __ffs(mm) - 1;
      mm &= mm - 1;
      s_idx[off++] = tid + i * THREADS;
    }
  }
  __syncthreads();

  const int half = tid >> 7;       // split neighbor list in two
  const int k    = (tid >> 6) & 1; // batch index
  const int o    = tid & 63;       // output feature
  float acc = 0.f;
  if (cnt > 0) {
    const float invS0 = 1.f / rS0, invS1 = 1.f / rS1;
    for (int base = 0; base < cnt; base += 1024) {
      const int lim = min(1024, cnt - base);
      for (int e = tid; e < lim; e += THREADS) {   // weights computed once
        const int m = s_idx[base + e];
        s_w0[e] = __expf(fmaxf(w10 + Wh2[m], 0.f) - rM0) * invS0;
        s_w1[e] = __expf(fmaxf(w11 + Wh2[NN + m], 0.f) - rM1) * invS1;
      }
      __syncthreads();
      for (int e = half; e < lim; e += 2) {
        const int m = s_idx[base + e];
        const float wv = k ? s_w1[e] : s_w0[e];
        acc = fmaf(wv, Wh[((size_t)k * NN + m) * OUT_F + o], acc);
      }
      __syncthreads();
    }
  } else {
    // all entries masked -> softmax(all -9e15) is uniform 1/NN (matches jax)
    for (int m = half; m < NN; m += 2)
      acc += Wh[((size_t)k * NN + m) * OUT_F + o];
    acc *= (1.0f / NN);
  }
  __syncthreads();
  sM0[tid] = acc;            // reuse reduction scratch to merge the two halves
  __syncthreads();
  if (half == 0)
    Hp[((size_t)k * NN + n) * OUT_F + o] = acc + sM0[tid + 128];
}

// ---------------------------------------------------------------------------
// Kernel 4: out[k,n,:] = relu( sum_{m: adj[n,m]>0} h_prime[k,m,:] )
// (adj entries are exactly 1.0 on edges)
// ---------------------------------------------------------------------------
__global__ __launch_bounds__(THREADS) void k_spmm_out(const float* __restrict__ adj,
                                                      const float* __restrict__ Hp,
                                                      float* __restrict__ out) {
  __shared__ int   s_idx[NN];
  __shared__ int   s_off[THREADS];
  __shared__ int   s_cnt;
  __shared__ float s_red[THREADS];
  const int n = blockIdx.x;
  const int tid = threadIdx.x;
  const float* arow = adj + (size_t)n * NN;
  unsigned mask = 0;
#pragma unroll 4
  for (int i = 0; i < NN / THREADS; ++i)
    if (arow[tid + i * THREADS] > 0.f) mask |= (1u << i);
  s_off[tid] = __popc(mask);
  __syncthreads();
  if (tid == 0) {
    int run = 0;
    for (int i = 0; i < THREADS; ++i) { const int t = s_off[i]; s_off[i] = run; run += t; }
    s_cnt = run;
  }
  __syncthreads();
  {
    int off = s_off[tid];
    unsigned mm = mask;
    while (mm) {
      const int i = __ffs(mm) - 1;
      mm &= mm - 1;
      s_idx[off++] = tid + i * THREADS;
    }
  }
  __syncthreads();
  const int cnt = s_cnt;
  const int half = tid >> 7, k = (tid >> 6) & 1, o = tid & 63;
  float acc = 0.f;
  for (int e = half; e < cnt; e += 2)
    acc += Hp[((size_t)k * NN + s_idx[e]) * OUT_F + o];
  s_red[tid] = acc;
  __syncthreads();
  if (half == 0)
    out[((size_t)k * NN + n) * OUT_F + o] = fmaxf(acc + s_red[tid + 128], 0.f);
}

// ---------------------------------------------------------------------------
extern "C" void kernel_launch(void* const* d_in, const int* in_sizes, int n_in,
                              void* d_out, int out_size, void* d_ws, size_t ws_size,
                              hipStream_t stream) {
  const float* x   = (const float*)d_in[0];   // [K,N,IN_F]
  const float* adj = (const float*)d_in[1];   // [N,N]
  const float* w   = (const float*)d_in[2];   // [IN_F,OUT_F]
  const float* a   = (const float*)d_in[3];   // [2*OUT_F,1]
  float* out = (float*)d_out;                 // [K,N,OUT_F]

  float* Wh  = (float*)d_ws;                        // K*N*OUT_F floats (4 MB)
  float* Hp  = Wh  + (size_t)KK * NN * OUT_F;       // K*N*OUT_F floats (4 MB)
  float* Wh1 = Hp  + (size_t)KK * NN * OUT_F;       // K*N floats
  float* Wh2 = Wh1 + (size_t)KK * NN;               // K*N floats

  dim3 g1(OUT_F / 16, (KK * NN) / 16);
  k_wh_wmma<<<g1, 32, 0, stream>>>(x, w, Wh);
  k_wh12<<<(KK * NN) / THREADS, THREADS, 0, stream>>>(Wh, a, Wh1, Wh2);
  k_attn<<<NN, THREADS, 0, stream>>>(adj, Wh, Wh1, Wh2, Hp);
  k_spmm_out<<<NN, THREADS, 0, stream>>>(adj, Hp, out);
}